// GAT_22308060136206
// MI455X (gfx1250) — compile-verified
//
#include <hip/hip_runtime.h>
#include <hip/hip_bf16.h>
#include <math.h>

typedef __attribute__((ext_vector_type(2))) float v2f;
typedef __attribute__((ext_vector_type(8))) float v8f;

#define NEG_SLOPE 0.2f

// ---------------------------------------------------------------------------
// f32 GEMM via V_WMMA_F32_16X16X4_F32. D[M,Ncol] = A[M,K] @ B[K,Ncol], all
// row-major f32. One wave32 computes one 16x16 tile. M, Ncol multiples of 16.
// A-matrix layout (ISA 7.12.2, 32-bit A 16x4): lanes 0-15 hold M=lane with
// VGPR0=K+0, VGPR1=K+1; lanes 16-31 hold M=lane-16 with K+2, K+3.
// B (4x16) is the mirror; C/D: VGPR v -> M = v + (lane>=16 ? 8 : 0), N = lane&15.
// ---------------------------------------------------------------------------
__global__ void gemm_wmma_f32(const float* __restrict__ A,
                              const float* __restrict__ B,
                              float* __restrict__ D,
                              int M, int K, int Ncol) {
  int wave = (blockIdx.x * blockDim.x + threadIdx.x) >> 5;   // wave-uniform
  int lane = threadIdx.x & 31;
  int tilesN = Ncol >> 4;
  int nTiles = (M >> 4) * tilesN;
  if (wave >= nTiles) return;                                // uniform branch
  int tm = wave / tilesN;
  int tn = wave - tm * tilesN;

  int r  = lane & 15;          // row within A-tile / col within B,D-tile
  int kh = (lane >> 4) << 1;   // 0 for lanes 0-15, 2 for lanes 16-31

  const float* Arow = A + (size_t)(tm * 16 + r) * K;
  const float* Bcol = B + (tn * 16 + r);

  v8f c = {};
  for (int k = 0; k < K; k += 4) {
    v2f a, b;
    a.x = Arow[k + kh];
    a.y = Arow[k + kh + 1];
    b.x = Bcol[(size_t)(k + kh) * Ncol];
    b.y = Bcol[(size_t)(k + kh + 1) * Ncol];
    // 8 args: (neg_a, A, neg_b, B, c_mod, C, reuse_a, reuse_b)
    c = __builtin_amdgcn_wmma_f32_16x16x4_f32(false, a, false, b,
                                              (short)0, c, false, false);
  }

  int col  = tn * 16 + r;
  int row0 = tm * 16 + ((lane >> 4) << 3);
#pragma unroll
  for (int v = 0; v < 8; ++v)
    D[(size_t)(row0 + v) * Ncol + col] = c[v];
}

// ---------------------------------------------------------------------------
// Per-(node, head) attention logits: al_src[n,h] = sum_c h[n,h,c]*a_src[h,c]
// ---------------------------------------------------------------------------
__global__ void attn_coef(const float* __restrict__ h,
                          const float* __restrict__ a_src,
                          const float* __restrict__ a_dst,
                          float* __restrict__ als, float* __restrict__ ald,
                          int n, int heads, int ch) {
  int t = blockIdx.x * blockDim.x + threadIdx.x;
  if (t >= n * heads) return;
  int node = t / heads;
  int hd   = t - node * heads;
  const float* hp = h + (size_t)node * heads * ch + hd * ch;
  const float* as = a_src + hd * ch;
  const float* ad = a_dst + hd * ch;
  float s = 0.f, d = 0.f;
  for (int c = 0; c < ch; ++c) {
    float v = hp[c];
    s = fmaf(v, as[c], s);
    d = fmaf(v, ad[c], d);
  }
  als[t] = s;
  ald[t] = d;
}

__device__ __forceinline__ float edge_weight(const float* als, const float* ald,
                                             int s, int d, int heads, int hd) {
  float x = als[s * heads + hd] + ald[d * heads + hd];
  float l = x > 0.f ? x : NEG_SLOPE * x;
  return expf(l);   // softmax shift-invariant; logits are O(1), no max needed
}

// ---------------------------------------------------------------------------
// Pass 1: denom[dst,h] += exp(leaky(al_src[src,h] + al_dst[dst,h]))
// Edges e < E come from edge_index; e >= E are self loops (node e-E).
// ---------------------------------------------------------------------------
__global__ void edge_denom(const int* __restrict__ ei, int E, int n, int heads,
                           const float* __restrict__ als,
                           const float* __restrict__ ald,
                           float* __restrict__ denom) {
  int t = blockIdx.x * blockDim.x + threadIdx.x;
  int total = (E + n) * heads;
  if (t >= total) return;
  int e  = t / heads;
  int hd = t - e * heads;
  int s  = (e < E) ? ei[e]     : (e - E);
  int d  = (e < E) ? ei[E + e] : (e - E);
  float w = edge_weight(als, ald, s, d, heads, hd);
  atomicAdd(&denom[d * heads + hd], w);
}

// ---------------------------------------------------------------------------
// Pass 2: out[dst,h,:] += (w / denom[dst,h]) * h[src,h,:]   (w recomputed)
// ---------------------------------------------------------------------------
__global__ void edge_agg(const int* __restrict__ ei, int E, int n, int heads,
                         int ch, const float* __restrict__ h,
                         const float* __restrict__ als,
                         const float* __restrict__ ald,
                         const float* __restrict__ denom,
                         float* __restrict__ out) {
  int t = blockIdx.x * blockDim.x + threadIdx.x;
  int total = (E + n) * heads;
  if (t >= total) return;
  int e  = t / heads;
  int hd = t - e * heads;
  int s  = (e < E) ? ei[e]     : (e - E);
  int d  = (e < E) ? ei[E + e] : (e - E);
  float w = edge_weight(als, ald, s, d, heads, hd);
  float alpha = w / (denom[d * heads + hd] + 1e-16f);
  const float* hs = h   + (size_t)s * heads * ch + hd * ch;
  float*       op = out + (size_t)d * heads * ch + hd * ch;
  for (int c = 0; c < ch; ++c)
    atomicAdd(&op[c], alpha * hs[c]);
}

// out[i] = elu(out[i] + bias[i % dim])   (in place)
__global__ void bias_elu(float* __restrict__ v, const float* __restrict__ b,
                         int n, int dim) {
  int t = blockIdx.x * blockDim.x + threadIdx.x;
  if (t >= n * dim) return;
  float x = v[t] + b[t % dim];
  v[t] = x > 0.f ? x : expm1f(x);
}

// d_out[n,:] = log_softmax(acc[n,:] + bias), 16 classes per node
__global__ void finalize_logsoftmax(const float* __restrict__ acc,
                                    const float* __restrict__ b,
                                    float* __restrict__ out, int n) {
  int t = blockIdx.x * blockDim.x + threadIdx.x;
  if (t >= n) return;
  float v[16];
  float m = -INFINITY;
#pragma unroll
  for (int c = 0; c < 16; ++c) {
    v[c] = acc[(size_t)t * 16 + c] + b[c];
    m = fmaxf(m, v[c]);
  }
  float sum = 0.f;
#pragma unroll
  for (int c = 0; c < 16; ++c) sum += expf(v[c] - m);
  float lse = m + logf(sum);
#pragma unroll
  for (int c = 0; c < 16; ++c)
    out[(size_t)t * 16 + c] = v[c] - lse;
}

extern "C" void kernel_launch(void* const* d_in, const int* in_sizes, int n_in,
                              void* d_out, int out_size, void* d_ws,
                              size_t ws_size, hipStream_t stream) {
  (void)n_in; (void)out_size; (void)ws_size;
  const float* x   = (const float*)d_in[0];
  const int*   ei  = (const int*)d_in[1];     // [2, E] int32 (JAX canonicalized)
  const float* W1  = (const float*)d_in[2];   // [128, 256]
  const float* a1s = (const float*)d_in[3];   // [8, 32]
  const float* a1d = (const float*)d_in[4];
  const float* b1  = (const float*)d_in[5];   // [256]
  const float* W2  = (const float*)d_in[6];   // [256, 16]
  const float* a2s = (const float*)d_in[7];   // [1, 16]
  const float* a2d = (const float*)d_in[8];
  const float* b2  = (const float*)d_in[9];   // [16]

  const int N  = in_sizes[0] / 128;           // 50000
  const int E  = in_sizes[1] / 2;             // 800000
  const int Ep = E + N;                       // with self loops

  // workspace layout (f32)
  float* ws   = (float*)d_ws;
  float* h1   = ws;                            // N*256  conv1 features
  float* act1 = h1   + (size_t)N * 256;        // N*256  aggregate -> ELU act
  float* l1s  = act1 + (size_t)N * 256;        // N*8
  float* l1d  = l1s  + (size_t)N * 8;          // N*8
  float* den1 = l1d  + (size_t)N * 8;          // N*8
  float* g2   = den1 + (size_t)N * 8;          // N*16   conv2 features
  float* l2s  = g2   + (size_t)N * 16;         // N
  float* l2d  = l2s  + (size_t)N;              // N
  float* den2 = l2d  + (size_t)N;              // N
  float* out2 = den2 + (size_t)N;              // N*16   conv2 aggregate

  // zero the accumulators (harness does not re-zero between replays)
  hipMemsetAsync(act1, 0, (size_t)N * 256 * sizeof(float), stream);
  hipMemsetAsync(den1, 0, (size_t)N * 8   * sizeof(float), stream);
  hipMemsetAsync(den2, 0, (size_t)N       * sizeof(float), stream);
  hipMemsetAsync(out2, 0, (size_t)N * 16  * sizeof(float), stream);

  const int TB = 256;

  // ---- layer 1 ----
  {
    int tiles = (N / 16) * (256 / 16);               // one wave per tile
    int blocks = (tiles + 3) / 4;                    // 4 waves / 128-thread block
    gemm_wmma_f32<<<blocks, 128, 0, stream>>>(x, W1, h1, N, 128, 256);
  }
  attn_coef<<<(N * 8 + TB - 1) / TB, TB, 0, stream>>>(h1, a1s, a1d, l1s, l1d, N, 8, 32);
  edge_denom<<<(Ep * 8 + TB - 1) / TB, TB, 0, stream>>>(ei, E, N, 8, l1s, l1d, den1);
  edge_agg<<<(Ep * 8 + TB - 1) / TB, TB, 0, stream>>>(ei, E, N, 8, 32, h1, l1s, l1d, den1, act1);
  bias_elu<<<((size_t)N * 256 + TB - 1) / TB, TB, 0, stream>>>(act1, b1, N, 256);

  // ---- layer 2 ----
  {
    int tiles = (N / 16) * (16 / 16);
    int blocks = (tiles + 3) / 4;
    gemm_wmma_f32<<<blocks, 128, 0, stream>>>(act1, W2, g2, N, 256, 16);
  }
  attn_coef<<<(N + TB - 1) / TB, TB, 0, stream>>>(g2, a2s, a2d, l2s, l2d, N, 1, 16);
  edge_denom<<<(Ep + TB - 1) / TB, TB, 0, stream>>>(ei, E, N, 1, l2s, l2d, den2);
  edge_agg<<<(Ep + TB - 1) / TB, TB, 0, stream>>>(ei, E, N, 1, 16, g2, l2s, l2d, den2, out2);
  finalize_logsoftmax<<<(N + TB - 1) / TB, TB, 0, stream>>>(out2, b2, (float*)d_out, N);
}